// GeneticFitness_13116830122112
// MI455X (gfx1250) — compile-verified
//
#include <hip/hip_runtime.h>
#include <math.h>

// Problem constants (match reference)
#define TT      1024
#define DD      64
#define BH      32          // B*H
#define WIN     128
#define NTILES  (TT / 16)   // 64 row tiles per head
#define XPAD    68          // padded LDS row stride in floats (mult of 4 -> 16B-aligned b128; (4r+c)%64 banks conflict-free)
#define MAXROWS 144
#define MAXCOLS 144

typedef float v2f __attribute__((ext_vector_type(2)));
typedef float v8f __attribute__((ext_vector_type(8)));

// ---------------------------------------------------------------------------
// Kernel 1: per 16-row tile, compute banded corr via V_WMMA_F32_16X16X4_F32
// and reduce to denom_i = 1 + sum_window(corr_raw)/2048   (exact reformulation
// of reference gene_means/denom since masked entries rescale to constant 0.5)
// ---------------------------------------------------------------------------
__global__ __launch_bounds__(128) void band_denom_kernel(
    const float* __restrict__ X, float* __restrict__ denom_out) {
  const int it   = blockIdx.x;           // row tile in head
  const int head = blockIdx.y;           // (b*H + h)
  const int i0   = it * 16;
  const int jt_begin = (it - 8) > 0 ? (it - 8) : 0;
  const int NT   = it - jt_begin + 1;    // 1..9 column tiles
  const int lo   = jt_begin * 16;        // first band row/col (global index)
  const int nrows = NT * 16;             // band rows == band cols (<=144)

  __shared__ float sX[MAXROWS * XPAD];
  __shared__ float snrm[MAXROWS];
  __shared__ float sbb[16 * MAXCOLS];

  const float* Xh = X + (size_t)head * TT * DD;
  const int tid = threadIdx.x;

  // stage band rows of X into LDS (float4, 16B-aligned since XPAD%4==0)
  const int nvec = nrows * (DD / 4);
  for (int v = tid; v < nvec; v += 128) {
    const int r = v >> 4, c4 = (v & 15) << 2;
    float4 val = *(const float4*)(Xh + (size_t)(lo + r) * DD + c4);
    *(float4*)(&sX[r * XPAD + c4]) = val;
  }
  for (int v = tid; v < 16 * MAXCOLS; v += 128) sbb[v] = 0.0f;
  __syncthreads();

  // per-row L2 norms from LDS
  for (int r = tid; r < nrows; r += 128) {
    float s = 0.0f;
#pragma unroll
    for (int c = 0; c < DD; ++c) { const float x = sX[r * XPAD + c]; s += x * x; }
    snrm[r] = sqrtf(s);
  }
  __syncthreads();

  const int wave = tid >> 5;
  const int lane = tid & 31;
  const int hf   = lane >> 4;       // which 16-lane half
  const int lr   = lane & 15;
  const int aBase = i0 - lo;        // band-row index of A-tile row 0

  for (int ct = wave; ct < NT; ct += 4) {
    const int bBase = ct * 16;      // band-row index of B-tile row 0
    v8f acc = {};
#pragma unroll
    for (int s = 0; s < 16; ++s) {
      const int k = 4 * s + 2 * hf;
      v2f a, b;
      a.x = sX[(aBase + lr) * XPAD + k];
      a.y = sX[(aBase + lr) * XPAD + k + 1];
      b.x = sX[(bBase + lr) * XPAD + k];
      b.y = sX[(bBase + lr) * XPAD + k + 1];
      acc = __builtin_amdgcn_wmma_f32_16x16x4_f32(false, a, false, b,
                                                  (short)0, acc, false, false);
    }
    const int j0 = lo + bBase;
#pragma unroll
    for (int v = 0; v < 8; ++v) {
      const int r = v + 8 * hf;      // row within tile
      const int i = i0 + r;
      const int j = j0 + lr;
      if (j <= i && j >= i - WIN) {
        const float ni = snrm[aBase + r];
        const float nj = snrm[bBase + lr];
        const float corr = acc[v] / (ni * nj + 1e-8f);
        sbb[r * MAXCOLS + (j - lo)] = corr;   // unique (r,j) per lane: no races
      }
    }
  }
  __syncthreads();

  // deterministic serial row reduction
  if (tid < 16) {
    float s = 0.0f;
    for (int c = 0; c < nrows; ++c) s += sbb[tid * MAXCOLS + c];
    denom_out[head * TT + i0 + tid] = 1.0f + s * (1.0f / 2048.0f);
  }
}

// ---------------------------------------------------------------------------
// Kernel 2: per head, S = sum_i 1/denom_i (deterministic tree reduction)
// ---------------------------------------------------------------------------
__global__ __launch_bounds__(256) void suminv_kernel(
    const float* __restrict__ denom, float* __restrict__ sum_inv) {
  const int head = blockIdx.x;
  __shared__ float red[256];
  float s = 0.0f;
  for (int t = threadIdx.x; t < TT; t += 256) s += 1.0f / denom[head * TT + t];
  red[threadIdx.x] = s;
  __syncthreads();
  for (int off = 128; off > 0; off >>= 1) {
    if ((int)threadIdx.x < off) red[threadIdx.x] += red[threadIdx.x + off];
    __syncthreads();
  }
  if (threadIdx.x == 0) sum_inv[head] = red[0];
}

// ---------------------------------------------------------------------------
// Kernel 3: recompute band, apply fitness, exp, row-normalize, stream out
// full 1024-wide rows (zeros outside window, matching exp(-1e10)->0).
// ---------------------------------------------------------------------------
__global__ __launch_bounds__(128) void band_softmax_kernel(
    const float* __restrict__ X, const float* __restrict__ denom,
    const float* __restrict__ sum_inv, float* __restrict__ out) {
  const int it   = blockIdx.x;
  const int head = blockIdx.y;
  const int i0   = it * 16;
  const int jt_begin = (it - 8) > 0 ? (it - 8) : 0;
  const int NT   = it - jt_begin + 1;
  const int lo   = jt_begin * 16;
  const int nrows = NT * 16;

  __shared__ float sX[MAXROWS * XPAD];
  __shared__ float snrm[MAXROWS];
  __shared__ float sbb[16 * MAXCOLS];
  __shared__ float sfit[16];
  __shared__ float sinvZ[16];

  const float* Xh = X + (size_t)head * TT * DD;
  const int tid = threadIdx.x;

  const int nvec = nrows * (DD / 4);
  for (int v = tid; v < nvec; v += 128) {
    const int r = v >> 4, c4 = (v & 15) << 2;
    float4 val = *(const float4*)(Xh + (size_t)(lo + r) * DD + c4);
    *(float4*)(&sX[r * XPAD + c4]) = val;
  }
  for (int v = tid; v < 16 * MAXCOLS; v += 128) sbb[v] = 0.0f;
  __syncthreads();

  for (int r = tid; r < nrows; r += 128) {
    float s = 0.0f;
#pragma unroll
    for (int c = 0; c < DD; ++c) { const float x = sX[r * XPAD + c]; s += x * x; }
    snrm[r] = sqrtf(s);
  }
  if (tid < 16) {
    const float S = sum_inv[head];
    sfit[tid] = 1.0f / (denom[head * TT + i0 + tid] * S);
  }
  __syncthreads();

  const int wave = tid >> 5;
  const int lane = tid & 31;
  const int hf   = lane >> 4;
  const int lr   = lane & 15;
  const int aBase = i0 - lo;

  for (int ct = wave; ct < NT; ct += 4) {
    const int bBase = ct * 16;
    v8f acc = {};
#pragma unroll
    for (int s = 0; s < 16; ++s) {
      const int k = 4 * s + 2 * hf;
      v2f a, b;
      a.x = sX[(aBase + lr) * XPAD + k];
      a.y = sX[(aBase + lr) * XPAD + k + 1];
      b.x = sX[(bBase + lr) * XPAD + k];
      b.y = sX[(bBase + lr) * XPAD + k + 1];
      acc = __builtin_amdgcn_wmma_f32_16x16x4_f32(false, a, false, b,
                                                  (short)0, acc, false, false);
    }
    const int j0 = lo + bBase;
#pragma unroll
    for (int v = 0; v < 8; ++v) {
      const int r = v + 8 * hf;
      const int i = i0 + r;
      const int j = j0 + lr;
      if (j <= i && j >= i - WIN) {
        const float ni = snrm[aBase + r];
        const float nj = snrm[bBase + lr];
        const float corr = acc[v] / (ni * nj + 1e-8f);
        const float w = (corr * 0.5f + 0.5f) * sfit[r];   // weighted, in (0, ~1e-3]
        sbb[r * MAXCOLS + (j - lo)] = __expf(w);
      }
    }
  }
  __syncthreads();

  if (tid < 16) {
    float z = 0.0f;
    for (int c = 0; c < nrows; ++c) z += sbb[tid * MAXCOLS + c];
    sinvZ[tid] = 1.0f / z;
  }
  __syncthreads();

  // stream full rows: 16 rows x 1024 cols, 8 floats per thread per row
  float* obase = out + ((size_t)head * TT + i0) * TT;
  const int cbase = tid * 8;
  for (int r = 0; r < 16; ++r) {
    const float invZ = sinvZ[r];
    float* orow = obase + (size_t)r * TT;
    float f[8];
#pragma unroll
    for (int q = 0; q < 8; ++q) {
      const int bi = cbase + q - lo;
      f[q] = (bi >= 0 && bi < nrows) ? sbb[r * MAXCOLS + bi] * invZ : 0.0f;
    }
    *(float4*)(orow + cbase)     = make_float4(f[0], f[1], f[2], f[3]);
    *(float4*)(orow + cbase + 4) = make_float4(f[4], f[5], f[6], f[7]);
  }
}

// ---------------------------------------------------------------------------
extern "C" void kernel_launch(void* const* d_in, const int* in_sizes, int n_in,
                              void* d_out, int out_size, void* d_ws, size_t ws_size,
                              hipStream_t stream) {
  (void)in_sizes; (void)n_in; (void)out_size; (void)ws_size;
  const float* X = (const float*)d_in[0];
  float* out     = (float*)d_out;
  float* denom   = (float*)d_ws;           // BH*T floats
  float* sum_inv = denom + BH * TT;        // BH floats

  dim3 grid(NTILES, BH);
  band_denom_kernel<<<grid, 128, 0, stream>>>(X, denom);
  suminv_kernel<<<BH, 256, 0, stream>>>(denom, sum_inv);
  band_softmax_kernel<<<grid, 128, 0, stream>>>(X, denom, sum_inv, out);
}